// GCN5_KHop_36197984370757
// MI455X (gfx1250) — compile-verified
//
#include <hip/hip_runtime.h>

typedef __attribute__((ext_vector_type(2))) float v2f;
typedef __attribute__((ext_vector_type(8))) float v8f;

#define N_NODES 10000
#define N_EDGES 16384
#define KHID    128
#define KSPLIT  2

// ---------------------------------------------------------------------------
// CDNA5 async helpers (gfx1250): per-lane 16B global->LDS copy, ASYNCcnt wait.
// ---------------------------------------------------------------------------
__device__ __forceinline__ void async_cp16(unsigned ldsOff, const void* g)
{
    asm volatile("global_load_async_to_lds_b128 %0, %1, off"
                 :: "v"(ldsOff), "v"(g) : "memory");
}
__device__ __forceinline__ void wait_async0()
{
    asm volatile("s_wait_asynccnt 0" ::: "memory");
}
// one software-pipeline stage: DSN next-step DS-reads, then WN current-step
// WMMAs (immediate constants required by the builtin).
template<int DSN, int WN>
__device__ __forceinline__ void sched_stage()
{
#if __has_builtin(__builtin_amdgcn_sched_group_barrier)
    __builtin_amdgcn_sched_group_barrier(0x100, DSN, 0);  // DS read
    __builtin_amdgcn_sched_group_barrier(0x008, WN, 0);   // MFMA/WMMA
#endif
}

// ---------------------------------------------------------------------------
// Per-layer W2 transform: pair-interleave the c' rows of each k-slice so a
// WMMA B fragment {row q, row q+1} at one column is a single aligned b64 LDS
// load. Virtual slice k=128 holds b2 (bias folded into the GEMM).
// ---------------------------------------------------------------------------
template<int C>
__global__ void w2t_kernel(const float* __restrict__ W2, const float* __restrict__ b2,
                           float* __restrict__ w2t)
{
    int idx = blockIdx.x * blockDim.x + threadIdx.x;
    if (idx >= (KHID + 1) * C * C) return;
    int k   = idx / (C * C);
    int rem = idx % (C * C);
    int c   = rem / C, d = rem % C;
    float v = (k < KHID) ? W2[idx] : b2[rem];
    w2t[(size_t)k * C * C + (((c >> 1) * C + d) * 2 + (c & 1))] = v;
}

// ---------------------------------------------------------------------------
// Heavy kernel: fused edge-message GEMM with f32 WMMA.
// msg[e,d] = sum_{k,c'} h[e,k]*x_src[e,c']*W2[k, c'*C+d]   (b2 folded as k=128)
// Wave tile: MS M-tiles x NT2 N-tiles (2x(C/32) for C>=32) so each B fragment feeds
// MS WMMAs. B/x software-pipelined one step ahead with a {DS,WMMA} group pair
// per step; W2 double-buffered via async-to-LDS; D scattered with f32 atomics.
// ---------------------------------------------------------------------------
template<int C>
__global__ __launch_bounds__(256) void msg_kernel(
    const float* __restrict__ hp,     // [N, C]
    const float* __restrict__ hedge,  // [E, 128]
    const float* __restrict__ w2t,    // [129][C*C] pair-interleaved slices
    const int*   __restrict__ ei,     // [2, E]
    float*       __restrict__ agg)    // [N, C]
{
    constexpr int MB   = 128;                                // edges per workgroup
    constexpr int NT   = C / 16;                             // 16x16 N-tiles total
    constexpr int MS   = (NT >= 2) ? 2 : 1;                  // M-tiles per wave
    constexpr int NT2  = NT / MS;                            // N-tiles per wave
    constexpr int CK   = (C >= 128) ? 1 : (16384 / (C * C)); // k-slices per 64KB chunk
    constexpr int KEFF = KHID + 1;
    constexpr int NCH  = (KEFF + CK - 1) / CK;
    // DS reads per step: NT2/2 B-pair loads (2addr_b64) + MS x-pair loads
    constexpr int DSN  = MS + ((NT2 >= 2) ? NT2 / 2 : 1);
    constexpr int WN   = MS * NT2;

    extern __shared__ float lds[];
    float* xs   = lds;                     // [MB][C]
    float* hs   = xs + MB * C;             // [MB][128]
    int*   sd   = (int*)(hs + MB * KHID);  // [2][MB]
    float* w2s0 = (float*)(sd + 2 * MB);   // [2][CK*C*C] double buffer
    const unsigned w2sOff0 = (unsigned)((const char*)w2s0 - (const char*)lds);
    constexpr unsigned CHB = (unsigned)CK * C * C * 4;       // chunk bytes

    const int tid  = threadIdx.x;
    const int lane = tid & 31;
    const int wave = tid >> 5;             // 0..7
    const int lid  = lane & 15;
    const int half = lane >> 4;            // 0 or 1
    const int eBase = blockIdx.x * MB;

    // wave -> (first M-tile, N half)
    const int mtb  = (MS == 2) ? ((wave >> 1) << 1) : wave;
    const int nh   = (MS == 2) ? (wave & 1) : 0;
    const int colB = nh * NT2 * 16;

    // this workgroup's chunk range (K-split over gridDim.y)
    const int cbeg = (NCH * blockIdx.y) / KSPLIT;
    const int cend = (NCH * (blockIdx.y + 1)) / KSPLIT;

    // kick off async staging of chunk cbeg into buffer 0
    {
        const int k0   = cbeg * CK;
        const int kend = (k0 + CK < KEFF) ? (k0 + CK) : KEFF;
        const char* src = (const char*)(w2t + (size_t)k0 * C * C);
        const int n16 = (kend - k0) * C * C / 4;
        for (int i = tid; i < n16; i += 256)
            async_cp16(w2sOff0 + (unsigned)i * 16, src + (size_t)i * 16);
    }
    for (int i = tid; i < MB; i += 256) {
        sd[i]      = ei[eBase + i];            // src
        sd[MB + i] = ei[N_EDGES + eBase + i];  // dst
    }
    __syncthreads();
    for (int i = tid; i < MB * C; i += 256) {
        int e = i / C, c = i % C;
        xs[i] = hp[sd[e] * C + c];
    }
    for (int i = tid; i < MB * KHID; i += 256) {
        hs[i] = hedge[(size_t)eBase * KHID + i];
    }
    wait_async0();
    __syncthreads();

    v8f zero = {};
    v8f acc[MS][NT2];
#pragma unroll
    for (int i = 0; i < MS; ++i)
#pragma unroll
        for (int t = 0; t < NT2; ++t) acc[i][t] = zero;

    int row[MS];
#pragma unroll
    for (int i = 0; i < MS; ++i) row[i] = (mtb + i) * 16 + lid;

    for (int ci = cbeg; ci < cend; ++ci) {
        const int k0     = ci * CK;
        const int kend   = (k0 + CK < KEFF) ? (k0 + CK) : KEFF;
        const int bufOff = ((ci - cbeg) & 1) * (CK * C * C);   // integer select

        // prefetch next chunk into the other buffer (overlapped with compute)
        if (ci + 1 < cend) {
            const int nk0   = (ci + 1) * CK;
            const int nkend = (nk0 + CK < KEFF) ? (nk0 + CK) : KEFF;
            const unsigned dstOff = w2sOff0 + (unsigned)((ci + 1 - cbeg) & 1) * CHB;
            const char* src = (const char*)(w2t + (size_t)nk0 * C * C);
            const int n16 = (nkend - nk0) * C * C / 4;
            for (int i = tid; i < n16; i += 256)
                async_cp16(dstOff + (unsigned)i * 16, src + (size_t)i * 16);
        }

        for (int k = k0; k < kend; ++k) {
            float hv[MS];
#pragma unroll
            for (int i = 0; i < MS; ++i)
                hv[i] = (k < KHID) ? hs[row[i] * KHID + k] : 1.0f;
            const float* w2r = w2s0 + bufOff + (size_t)(k - k0) * C * C;
            const float* bpk = w2r + (size_t)half * (2 * C) + (colB + lid) * 2;

            // software pipeline: next step's B/x loads issue before this
            // step's WMMAs; one {DS,WMMA} group pair per step keeps the
            // one-stage distance so waits stay at dscnt<=DSN.
            v2f xp[MS];
#pragma unroll
            for (int i = 0; i < MS; ++i)
                xp[i] = *(const v2f*)(xs + row[i] * C + 2 * half);
            v2f Bcur[NT2];
#pragma unroll
            for (int t = 0; t < NT2; ++t) Bcur[t] = *(const v2f*)(bpk + t * 32);

#pragma unroll 4
            for (int c0 = 0; c0 < C - 4; c0 += 4) {
                v2f A[MS];
#pragma unroll
                for (int i = 0; i < MS; ++i) A[i] = hv[i] * xp[i];   // v_pk_mul_f32
#pragma unroll
                for (int i = 0; i < MS; ++i)
                    xp[i] = *(const v2f*)(xs + row[i] * C + (c0 + 4) + 2 * half);
                v2f Bnxt[NT2];
                const float* bpn = bpk + (size_t)(c0 / 2 + 2) * (2 * C);
#pragma unroll
                for (int t = 0; t < NT2; ++t) Bnxt[t] = *(const v2f*)(bpn + t * 32);
#pragma unroll
                for (int i = 0; i < MS; ++i)
#pragma unroll
                    for (int t = 0; t < NT2; ++t)
                        acc[i][t] = __builtin_amdgcn_wmma_f32_16x16x4_f32(
                            false, A[i], false, Bcur[t], (short)0, acc[i][t], false, false);
                sched_stage<DSN, WN>();
#pragma unroll
                for (int t = 0; t < NT2; ++t) Bcur[t] = Bnxt[t];
            }
            {   // peeled last step (c0 = C-4)
                v2f A[MS];
#pragma unroll
                for (int i = 0; i < MS; ++i) A[i] = hv[i] * xp[i];
#pragma unroll
                for (int i = 0; i < MS; ++i)
#pragma unroll
                    for (int t = 0; t < NT2; ++t)
                        acc[i][t] = __builtin_amdgcn_wmma_f32_16x16x4_f32(
                            false, A[i], false, Bcur[t], (short)0, acc[i][t], false, false);
            }
        }
        wait_async0();
        __syncthreads();
    }

    // D layout: acc[i][t][j] is element (edge (mtb+i)*16 + j + 8*half,
    //                                    col  colB + t*16 + lid)
#pragma unroll
    for (int i = 0; i < MS; ++i) {
#pragma unroll
        for (int j = 0; j < 8; ++j) {
            const int er  = (mtb + i) * 16 + j + 8 * half;
            const int dst = sd[MB + er];
#pragma unroll
            for (int t = 0; t < NT2; ++t) {
                atomicAdd(&agg[(size_t)dst * C + colB + t * 16 + lid], acc[i][t][j]);
            }
        }
    }
}

// ---------------------------------------------------------------------------
// Cheap VALU kernels (<2% of total FLOPs)
// ---------------------------------------------------------------------------
__global__ void proj_kernel(const float* __restrict__ hin, const float* __restrict__ W,
                            const float* __restrict__ b, float* __restrict__ hout,
                            int cin, int cout)
{
    int idx = blockIdx.x * blockDim.x + threadIdx.x;
    if (idx >= N_NODES * cout) return;
    int n = idx / cout, d = idx % cout;
    float s = b[d];
    const float* hr = hin + (size_t)n * cin;
    for (int i = 0; i < cin; ++i) s += hr[i] * W[(size_t)i * cout + d];
    hout[idx] = s;
}

__global__ void edge_hidden_kernel(const float* __restrict__ ea, const float* __restrict__ W1,
                                   const float* __restrict__ b1, float* __restrict__ he)
{
    int idx = blockIdx.x * blockDim.x + threadIdx.x;
    if (idx >= N_EDGES * KHID) return;
    int e = idx >> 7, j = idx & (KHID - 1);
    he[idx] = fmaxf(ea[e] * W1[j] + b1[j], 0.0f);
}

__global__ void count_kernel(const int* __restrict__ ei, float* __restrict__ cnt)
{
    int e = blockIdx.x * blockDim.x + threadIdx.x;
    if (e < N_EDGES) atomicAdd(&cnt[ei[N_EDGES + e]], 1.0f);
}

template<int C>
__global__ void combine_kernel(const float* __restrict__ agg, const float* __restrict__ cnt,
                               const float* __restrict__ hp, const float* __restrict__ root,
                               const float* __restrict__ bias, float* __restrict__ hout)
{
    int idx = blockIdx.x * blockDim.x + threadIdx.x;
    if (idx >= N_NODES * C) return;
    int n = idx / C, d = idx % C;
    const float* hr = hp + (size_t)n * C;
    float s = bias[d];
#pragma unroll 8
    for (int i = 0; i < C; ++i) s += hr[i] * root[i * C + d];
    float v = agg[idx] / fmaxf(cnt[n], 1.0f) + s;
    hout[idx] = fmaxf(v, 0.0f) + hr[d];
}

__global__ void head_kernel(const float* __restrict__ h, const int* __restrict__ ei,
                            const float* __restrict__ W1, const float* __restrict__ b1,
                            const float* __restrict__ W2, const float* __restrict__ b2,
                            float* __restrict__ out)
{
    int e = blockIdx.x * blockDim.x + threadIdx.x;
    if (e >= N_EDGES) return;
    int r = ei[e], c = ei[N_EDGES + e];
    float v[16];
#pragma unroll
    for (int i = 0; i < 16; ++i) v[i] = h[r * 16 + i] + h[c * 16 + i];
    float o = b2[0];
#pragma unroll
    for (int j = 0; j < 16; ++j) {
        float t = b1[j];
#pragma unroll
        for (int i = 0; i < 16; ++i) t += v[i] * W1[i * 16 + j];
        o += fmaxf(t, 0.0f) * W2[j];
    }
    out[e] = o;
}

// ---------------------------------------------------------------------------
// Launch
// ---------------------------------------------------------------------------
template<int C>
static void run_layer(const float* hin, int cin, const float* ea, const int* ei,
                      const float* const* prm, float* hp, float* hedge, float* agg,
                      float* w2t, const float* cnt, float* hout, hipStream_t stream)
{
    constexpr int MB = 128;
    constexpr int CK = (C >= 128) ? 1 : (16384 / (C * C));
    proj_kernel<<<(N_NODES * C + 255) / 256, 256, 0, stream>>>(hin, prm[0], prm[1], hp, cin, C);
    edge_hidden_kernel<<<(N_EDGES * KHID + 255) / 256, 256, 0, stream>>>(ea, prm[2], prm[3], hedge);
    w2t_kernel<C><<<((KHID + 1) * C * C + 255) / 256, 256, 0, stream>>>(prm[4], prm[5], w2t);
    (void)hipMemsetAsync(agg, 0, (size_t)N_NODES * C * sizeof(float), stream);
    size_t lds = ((size_t)MB * C + (size_t)MB * KHID + 2 * MB + 2 * (size_t)CK * C * C) * 4;
    dim3 grid(N_EDGES / MB, KSPLIT);
    msg_kernel<C><<<grid, 256, lds, stream>>>(hp, hedge, w2t, ei, agg);
    combine_kernel<C><<<(N_NODES * C + 255) / 256, 256, 0, stream>>>(agg, cnt, hp, prm[6], prm[7], hout);
}

extern "C" void kernel_launch(void* const* d_in, const int* in_sizes, int n_in,
                              void* d_out, int out_size, void* d_ws, size_t ws_size,
                              hipStream_t stream)
{
    const float* x  = (const float*)d_in[0];
    const int*   ei = (const int*)  d_in[1];
    const float* ea = (const float*)d_in[2];
    const float* prm[4][8];
    for (int li = 0; li < 4; ++li)
        for (int j = 0; j < 8; ++j)
            prm[li][j] = (const float*)d_in[3 + li * 8 + j];
    const float* el1W = (const float*)d_in[35];
    const float* el1b = (const float*)d_in[36];
    const float* el2W = (const float*)d_in[37];
    const float* el2b = (const float*)d_in[38];

    float* ws    = (float*)d_ws;
    float* nodeA = ws;                               // [N,128]
    float* nodeB = nodeA + (size_t)N_NODES * 128;    // [N,128]
    float* hp    = nodeB + (size_t)N_NODES * 128;    // [N,128]
    float* agg   = hp    + (size_t)N_NODES * 128;    // [N,128]
    float* hedge = agg   + (size_t)N_NODES * 128;    // [E,128]
    float* cnt   = hedge + (size_t)N_EDGES * 128;    // [N]
    float* w2t   = cnt   + N_NODES;                  // [129*128*128] max
    float* out   = (float*)d_out;

    (void)hipMemsetAsync(cnt, 0, N_NODES * sizeof(float), stream);
    count_kernel<<<(N_EDGES + 255) / 256, 256, 0, stream>>>(ei, cnt);

    run_layer<128>(x,     1,   ea, ei, prm[0], hp, hedge, agg, w2t, cnt, nodeA, stream);
    run_layer<64 >(nodeA, 128, ea, ei, prm[1], hp, hedge, agg, w2t, cnt, nodeB, stream);
    run_layer<32 >(nodeB, 64,  ea, ei, prm[2], hp, hedge, agg, w2t, cnt, nodeA, stream);
    run_layer<16 >(nodeA, 32,  ea, ei, prm[3], hp, hedge, agg, w2t, cnt, nodeB, stream);

    head_kernel<<<(N_EDGES + 255) / 256, 256, 0, stream>>>(nodeB, ei, el1W, el1b, el2W, el2b, out);
}